// TradicGCN_33818572489412
// MI455X (gfx1250) — compile-verified
//
#include <hip/hip_runtime.h>
#include <math.h>

// Problem constants (match reference)
#define NN   50000
#define RRN  400
#define DD   300
#define EE   800000
#define BBN  2048
#define LLN  32
#define HHN  8
#define NTN  50400   // NN + RRN

// GEMM tiling
#define KC   128     // K chunk staged in LDS
#define SAST 132     // LDS row stride (KC+4): bank-conflict-free b64 reads (wave32)

typedef float v2f __attribute__((ext_vector_type(2)));
typedef float v8f __attribute__((ext_vector_type(8)));

__device__ __forceinline__ float tg_wave_sum(float v) {
#pragma unroll
  for (int o = 16; o > 0; o >>= 1) v += __shfl_xor(v, o, 32);
  return v;
}
__device__ __forceinline__ float tg_wave_max(float v) {
#pragma unroll
  for (int o = 16; o > 0; o >>= 1) v = fmaxf(v, __shfl_xor(v, o, 32));
  return v;
}

// ---------------------------------------------------------------- fill
__global__ void tg_fill_f32(float* __restrict__ p, float v, long long n) {
  long long i = (long long)blockIdx.x * blockDim.x + threadIdx.x;
  if (i < n) p[i] = v;
}

// ------------------------------------------------- edge scatter (segment_sum)
__global__ void tg_scatter_edges(const float* __restrict__ x,
                                 const float* __restrict__ w,
                                 const int* __restrict__ src,
                                 const int* __restrict__ dst,
                                 float* __restrict__ agg, long long total) {
  long long tid = (long long)blockIdx.x * blockDim.x + threadIdx.x;
  if (tid >= total) return;
  long long e = tid / 75;
  int c = (int)(tid % 75) * 4;
  float ww = w[e];
  const float4 xv = *(const float4*)(x + (size_t)src[e] * DD + c);
  float* out = agg + (size_t)dst[e] * DD + c;
  atomicAdd(out + 0, ww * xv.x);
  atomicAdd(out + 1, ww * xv.y);
  atomicAdd(out + 2, ww * xv.z);
  atomicAdd(out + 3, ww * xv.w);
}

// ------------------------------------------------- LDS-staged f32-WMMA GEMM
// C[M x Ncols] = op( gatherA @ B + bias )
// gatherA row r = A1[idx1[r]] (+ A2[idx2[r]] if HAS2)  (HASIDX=0 -> plain rows)
// Block: 256 thr = 8 waves, tile 32(M) x 64(N); per-wave 16x16 C tile.
// K staged in LDS chunks of KC: sA[32][SAST] row-major, sB[64][SAST] = B^T panel.
template <int HASIDX, int HAS2>
__global__ __launch_bounds__(256) void tg_wmma_gemm(
    const float* __restrict__ A1, const int* __restrict__ idx1,
    const float* __restrict__ A2, const int* __restrict__ idx2,
    int lda,
    const float* __restrict__ Bm, int ldb,
    const float* __restrict__ bias,
    float* __restrict__ C, int ldc,
    int M, int K, int Ncols, int relu) {
  __shared__ float sA[32 * SAST];
  __shared__ float sB[64 * SAST];
  int tid  = threadIdx.x;
  int wave = tid >> 5;
  int lane = tid & 31;
  int half = lane >> 4;
  int l16  = lane & 15;
  int mBlk = blockIdx.y * 32;
  int m0   = mBlk + (wave >> 2) * 16;
  int n0   = blockIdx.x * 64;
  int cw   = (wave & 3) * 16;          // wave col offset inside block tile
  int col  = n0 + cw + l16;
  int rowL = (wave >> 2) * 16 + l16;   // local A row 0..31
  int koff = half * 2;                 // lanes 16-31 carry K+2,K+3 (ISA 16x4 A layout)

  v8f c = {0.f, 0.f, 0.f, 0.f, 0.f, 0.f, 0.f, 0.f};

  for (int k0 = 0; k0 < K; k0 += KC) {
    int kLen  = (K - k0 < KC) ? (K - k0) : KC;   // always even (K % 4 == 0)
    int half2 = kLen >> 1;
    // ---- stage A (32 x kLen), float2 granularity, gather+sum fused
    for (int i = tid; i < 32 * half2; i += 256) {
      int r = i / half2;
      int p = (i - r * half2) * 2;
      int gr = mBlk + r;
      float2 v = make_float2(0.f, 0.f);
      if (gr < M) {
        const float* ap =
            A1 + (size_t)(HASIDX ? idx1[gr] : gr) * lda + k0 + p;
        v = *(const float2*)ap;
        if (HAS2) {
          const float* ap2 = A2 + (size_t)idx2[gr] * lda + k0 + p;
          float2 v2 = *(const float2*)ap2;
          v.x += v2.x;
          v.y += v2.y;
        }
      }
      *(float2*)&sA[r * SAST + p] = v;
    }
    // ---- stage B^T (64 cols x kLen), coalesced global reads per k row
    for (int i = tid; i < 64 * kLen; i += 256) {
      int kk = i >> 6;
      int cc = i & 63;
      int gc = n0 + cc;
      float v = (gc < Ncols) ? Bm[(size_t)(k0 + kk) * ldb + gc] : 0.f;
      sB[cc * SAST + kk] = v;
    }
    __syncthreads();
    // ---- branch-free WMMA loop: 2x ds_load_b64 + wmma per step
    const float* pa = &sA[rowL * SAST + koff];
    const float* pb = &sB[(cw + l16) * SAST + koff];
    for (int kk = 0; kk < kLen; kk += 4) {
      v2f a = *(const v2f*)(pa + kk);
      v2f b = *(const v2f*)(pb + kk);
      c = __builtin_amdgcn_wmma_f32_16x16x4_f32(false, a, false, b, (short)0, c,
                                                false, false);
    }
    __syncthreads();
  }

  if (m0 < M && col < Ncols) {
#pragma unroll
    for (int vv = 0; vv < 8; ++vv) {
      int r = m0 + vv + half * 8;      // C layout: VGPR v -> M=v / v+8
      float val = c[vv];
      if (bias) val += bias[col];
      if (relu) val = fmaxf(val, 0.f);
      C[(size_t)r * ldc + col] = val;
    }
  }
}

// ------------------------------------------------- attention (block = row b)
// 8 wave32 = 8 heads; lane = DK dim (DK=32)
__global__ __launch_bounds__(256) void tg_attention(
    const float* __restrict__ q, const float* __restrict__ k,
    const float* __restrict__ v, const unsigned char* __restrict__ mask,
    float* __restrict__ ctx) {
  int b = blockIdx.x;
  int h = threadIdx.x >> 5;
  int lane = threadIdx.x & 31;
  size_t qoff = (size_t)b * 256 + h * 32 + lane;
  float qv = q[qoff];
  float myscore = 0.f;
#pragma unroll 4
  for (int l = 0; l < LLN; ++l) {
    float kv = k[(((size_t)b * LLN + l) * 256) + h * 32 + lane];
    float p = tg_wave_sum(qv * kv);
    if (lane == l) myscore = p;
  }
  bool m = mask[(size_t)b * LLN + lane] != 0;
  float any = tg_wave_max(m ? 1.f : 0.f);
  myscore = m ? myscore * 0.17677669529663687f /* 1/sqrt(32) */ : -1.0e9f;
  float mx = tg_wave_max(myscore);
  float e = __expf(myscore - mx);
  float s = tg_wave_sum(e);
  float attn = e / s;
  float acc = 0.f;
#pragma unroll 4
  for (int l = 0; l < LLN; ++l) {
    float al = __shfl(attn, l, 32);
    acc += al * v[(((size_t)b * LLN + l) * 256) + h * 32 + lane];
  }
  ctx[qoff] = (any > 0.f) ? acc : 0.f;   // where(mask.any(), out, 0)
}

// ------------------------------------------------- gather rows: dst[b,:300]=src[ids[b]]
__global__ void tg_gather_rows(const float* __restrict__ src,
                               const int* __restrict__ ids,
                               float* __restrict__ dst, int ldd, long long total) {
  long long t = (long long)blockIdx.x * blockDim.x + threadIdx.x;
  if (t >= total) return;
  int b = (int)(t / DD);
  int c = (int)(t % DD);
  dst[(size_t)b * ldd + c] = src[(size_t)ids[b] * DD + c];
}

// ------------------------------------------------- row L2 normalize (1 wave/row)
__global__ void tg_normalize_rows(const float* __restrict__ in,
                                  float* __restrict__ out, int cols) {
  int rowi = blockIdx.x;
  int lane = threadIdx.x;
  const float* ip = in + (size_t)rowi * cols;
  float s = 0.f;
  for (int c = lane; c < cols; c += 32) { float vv = ip[c]; s += vv * vv; }
  s = tg_wave_sum(s);
  float inv = 1.f / fmaxf(sqrtf(s), 1e-8f);
  float* op = out + (size_t)rowi * cols;
  for (int c = lane; c < cols; c += 32) op[c] = ip[c] * inv;
}

// ------------------------------------------------- fused sim (A@B^T) + margin loss
// Same LDS-staged WMMA structure; K=600, M=N=2048 (no guards needed).
__global__ __launch_bounds__(256) void tg_simloss(const float* __restrict__ lemn,
                                                  const float* __restrict__ remn,
                                                  float* __restrict__ accum) {
  __shared__ float sA[32 * SAST];
  __shared__ float sB[64 * SAST];
  int tid  = threadIdx.x;
  int wave = tid >> 5;
  int lane = tid & 31;
  int half = lane >> 4;
  int l16  = lane & 15;
  int mBlk = blockIdx.y * 32;
  int m0   = mBlk + (wave >> 2) * 16;
  int n0   = blockIdx.x * 64;
  int cw   = (wave & 3) * 16;
  int col  = n0 + cw + l16;
  int rowL = (wave >> 2) * 16 + l16;
  int koff = half * 2;

  v8f c = {0.f, 0.f, 0.f, 0.f, 0.f, 0.f, 0.f, 0.f};

  for (int k0 = 0; k0 < 600; k0 += KC) {
    int kLen  = (600 - k0 < KC) ? (600 - k0) : KC;
    int half2 = kLen >> 1;
    for (int i = tid; i < 32 * half2; i += 256) {
      int r = i / half2;
      int p = (i - r * half2) * 2;
      float2 v = *(const float2*)(lemn + (size_t)(mBlk + r) * 600 + k0 + p);
      *(float2*)&sA[r * SAST + p] = v;
    }
    // B^T panel: sB[cc][kk] = remn[(n0+cc)][k0+kk]  (remn rows ARE B columns)
    for (int i = tid; i < 64 * half2; i += 256) {
      int cc = i / half2;
      int p = (i - cc * half2) * 2;
      float2 v = *(const float2*)(remn + (size_t)(n0 + cc) * 600 + k0 + p);
      *(float2*)&sB[cc * SAST + p] = v;
    }
    __syncthreads();
    const float* pa = &sA[rowL * SAST + koff];
    const float* pb = &sB[(cw + l16) * SAST + koff];
    for (int kk = 0; kk < kLen; kk += 4) {
      v2f a = *(const v2f*)(pa + kk);
      v2f b = *(const v2f*)(pb + kk);
      c = __builtin_amdgcn_wmma_f32_16x16x4_f32(false, a, false, b, (short)0, c,
                                                false, false);
    }
    __syncthreads();
  }

  float nsum = 0.f, psum = 0.f;
#pragma unroll
  for (int vv = 0; vv < 8; ++vv) {
    int i = m0 + vv + half * 8;
    float val = c[vv];
    float e = (i == col) ? 0.f : val;        // no_diag
    nsum += fmaxf(e, 0.2f);                  // max(no_diag, GAMMA2)
    if (i == col) psum += fminf(val, 0.9f);  // min(diag, GAMMA1)
  }
  nsum = tg_wave_sum(nsum);
  psum = tg_wave_sum(psum);
  if (lane == 0) {
    atomicAdd(&accum[1], nsum);
    atomicAdd(&accum[0], psum);
  }
}

__global__ void tg_finalize(const float* __restrict__ accum, float* __restrict__ out) {
  out[0] = accum[1] / ((float)BBN * (float)BBN) - 0.2f - accum[0] / (float)BBN + 0.9f;
}

// ================================================================ launch
extern "C" void kernel_launch(void* const* d_in, const int* in_sizes, int n_in,
                              void* d_out_v, int out_size, void* d_ws, size_t ws_size,
                              hipStream_t stream) {
  const float* e_x    = (const float*)d_in[0];
  const float* r_x    = (const float*)d_in[1];
  const float* edge_w = (const float*)d_in[2];
  const float* Wrel   = (const float*)d_in[3];
  const float* brel   = (const float*)d_in[4];
  const float* W1     = (const float*)d_in[5];
  const float* b1     = (const float*)d_in[6];
  const float* W2     = (const float*)d_in[7];
  const float* b2     = (const float*)d_in[8];
  const float* Wprox  = (const float*)d_in[9];
  const float* bprox  = (const float*)d_in[10];
  const float* encWq  = (const float*)d_in[11];   // (2, 300, 256)
  const float* encWk  = (const float*)d_in[12];
  const float* encWv  = (const float*)d_in[13];
  const float* encWo  = (const float*)d_in[14];   // (2, 256, 300)
  const int* edge_src = (const int*)d_in[15];
  const int* edge_dst = (const int*)d_in[16];
  const int* ids_[2]  = {(const int*)d_in[17], (const int*)d_in[18]};
  const int* rel_[2][2]  = {{(const int*)d_in[19], (const int*)d_in[22]},
                            {(const int*)d_in[25], (const int*)d_in[28]}};
  const int* head_[2][2] = {{(const int*)d_in[20], (const int*)d_in[23]},
                            {(const int*)d_in[26], (const int*)d_in[29]}};
  const unsigned char* mask_[2][2] =
      {{(const unsigned char*)d_in[21], (const unsigned char*)d_in[24]},
       {(const unsigned char*)d_in[27], (const unsigned char*)d_in[30]}};
  float* d_out = (float*)d_out_v;   // [loss(1) | lem(2048*600) | rem(2048*600)]

  float* ws = (float*)d_ws;
  size_t off = 0;
  float* bufA  = ws + off; off += (size_t)NTN * DD;       // x -> h1 -> h2
  float* bufB  = ws + off; off += (size_t)NTN * DD;       // agg1 / agg2
  float* qbuf  = ws + off; off += (size_t)BBN * 256;
  float* kbuf  = ws + off; off += (size_t)BBN * LLN * 256;
  float* vbuf  = ws + off; off += (size_t)BBN * LLN * 256;
  float* ctxb  = ws + off; off += (size_t)BBN * 256;
  float* embuf = ws + off; off += (size_t)BBN * 600;
  float* lemn  = ws + off; off += (size_t)BBN * 600;
  float* remn  = ws + off; off += (size_t)BBN * 600;
  float* accum = ws + off; off += 16;

  auto gemm = [&](const float* A1, const int* i1, const float* A2, const int* i2,
                  int lda, const float* Bm, int ldb, const float* bias,
                  float* C, int ldc, int M, int K, int Ncols, int relu) {
    dim3 grid((Ncols + 63) / 64, (M + 31) / 32);
    if (A2)
      tg_wmma_gemm<1, 1><<<grid, 256, 0, stream>>>(A1, i1, A2, i2, lda, Bm, ldb,
                                                   bias, C, ldc, M, K, Ncols, relu);
    else if (i1)
      tg_wmma_gemm<1, 0><<<grid, 256, 0, stream>>>(A1, i1, A2, i2, lda, Bm, ldb,
                                                   bias, C, ldc, M, K, Ncols, relu);
    else
      tg_wmma_gemm<0, 0><<<grid, 256, 0, stream>>>(A1, i1, A2, i2, lda, Bm, ldb,
                                                   bias, C, ldc, M, K, Ncols, relu);
  };

  // 1) x = concat(e_x, r_x @ Wrel + brel)
  hipMemcpyAsync(bufA, e_x, (size_t)NN * DD * sizeof(float),
                 hipMemcpyDeviceToDevice, stream);
  gemm(r_x, nullptr, nullptr, nullptr, 2 * DD, Wrel, DD, brel,
       bufA + (size_t)NN * DD, DD, RRN, 2 * DD, DD, 0);

  // 2) agg1 = segment_sum(edge_w * x[src] -> dst)
  {
    long long nf = (long long)NTN * DD;
    tg_fill_f32<<<(unsigned)((nf + 255) / 256), 256, 0, stream>>>(bufB, 0.f, nf);
    long long total = (long long)EE * 75;
    tg_scatter_edges<<<(unsigned)((total + 255) / 256), 256, 0, stream>>>(
        bufA, edge_w, edge_src, edge_dst, bufB, total);
  }
  // 3) h1 = relu(agg1 @ W1 + b1)
  gemm(bufB, nullptr, nullptr, nullptr, DD, W1, DD, b1, bufA, DD, NTN, DD, DD, 1);
  // 4) agg2
  {
    long long nf = (long long)NTN * DD;
    tg_fill_f32<<<(unsigned)((nf + 255) / 256), 256, 0, stream>>>(bufB, 0.f, nf);
    long long total = (long long)EE * 75;
    tg_scatter_edges<<<(unsigned)((total + 255) / 256), 256, 0, stream>>>(
        bufA, edge_w, edge_src, edge_dst, bufB, total);
  }
  // 5) h2 = agg2 @ W2 + b2 -> bufA   (ent_f = h2[:N], rel_f = h2[N:])
  gemm(bufB, nullptr, nullptr, nullptr, DD, W2, DD, b2, bufA, DD, NTN, DD, DD, 0);

  const float* entf = bufA;
  const float* relf = bufA + (size_t)NN * DD;

  // 6) encoders + proximity for each side
  for (int side = 0; side < 2; ++side) {
    float* outp = d_out + 1 + (size_t)side * BBN * 600;
    for (int dir = 0; dir < 2; ++dir) {        // 0 = in, 1 = out
      const float* Wq = encWq + (size_t)dir * DD * 256;
      const float* Wk = encWk + (size_t)dir * DD * 256;
      const float* Wv = encWv + (size_t)dir * DD * 256;
      const float* Wo = encWo + (size_t)dir * 256 * DD;
      // q = ent_f[ids] @ Wq
      gemm(entf, ids_[side], nullptr, nullptr, DD, Wq, 256, nullptr,
           qbuf, 256, BBN, DD, 256, 0);
      // tok = ent_f[head] + rel_f[rel];  k = tok @ Wk;  v = tok @ Wv
      gemm(entf, head_[side][dir], relf, rel_[side][dir], DD, Wk, 256, nullptr,
           kbuf, 256, BBN * LLN, DD, 256, 0);
      gemm(entf, head_[side][dir], relf, rel_[side][dir], DD, Wv, 256, nullptr,
           vbuf, 256, BBN * LLN, DD, 256, 0);
      tg_attention<<<BBN, 256, 0, stream>>>(qbuf, kbuf, vbuf, mask_[side][dir], ctxb);
      // em[:, dir*300 : dir*300+300] = ctx @ Wo
      gemm(ctxb, nullptr, nullptr, nullptr, 256, Wo, DD, nullptr,
           embuf + dir * DD, 2 * DD, BBN, 256, DD, 0);
    }
    // prox = em @ Wprox + bprox -> out[:, 300:600]
    gemm(embuf, nullptr, nullptr, nullptr, 2 * DD, Wprox, DD, bprox,
         outp + DD, 2 * DD, BBN, 2 * DD, DD, 0);
    // out[:, 0:300] = kernel = ent_f[ids]
    long long total = (long long)BBN * DD;
    tg_gather_rows<<<(unsigned)((total + 255) / 256), 256, 0, stream>>>(
        entf, ids_[side], outp, 2 * DD, total);
  }

  // 7) cosine-sim loss
  tg_normalize_rows<<<BBN, 32, 0, stream>>>(d_out + 1, lemn, 600);
  tg_normalize_rows<<<BBN, 32, 0, stream>>>(d_out + 1 + (size_t)BBN * 600, remn, 600);
  tg_fill_f32<<<1, 32, 0, stream>>>(accum, 0.f, 2);
  tg_simloss<<<dim3(BBN / 64, BBN / 32), 256, 0, stream>>>(lemn, remn, accum);
  tg_finalize<<<1, 1, 0, stream>>>(accum, d_out);
}